// TraditionalGNN_9543417332444
// MI455X (gfx1250) — compile-verified
//
#include <hip/hip_runtime.h>
#include <math.h>

#define N_NODES   50000
#define N_EDGES   800000
#define DIM       128
#define NUM_G     64

typedef __attribute__((ext_vector_type(2))) float v2f;
typedef __attribute__((ext_vector_type(8))) float v8f;

// ---------------------------------------------------------------------------
// Degree / normalization kernels
// ---------------------------------------------------------------------------
__global__ void deg_init(float* deg, int n) {
    int i = blockIdx.x * blockDim.x + threadIdx.x;
    if (i < n) deg[i] = 1.0f;                 // self-loop contributes 1
}

__global__ void deg_count(const int* __restrict__ dst, float* deg, int e) {
    int i = blockIdx.x * blockDim.x + threadIdx.x;
    if (i < e) atomicAdd(&deg[dst[i]], 1.0f);
}

__global__ void deg_rsqrt(float* deg, int n) {
    int i = blockIdx.x * blockDim.x + threadIdx.x;
    if (i < n) deg[i] = rsqrtf(deg[i]);       // deg >= 1 always
}

// ---------------------------------------------------------------------------
// GEMM: P = (X @ W) * dinv[row]     X:[N,128] W:[128,128] row-major
// One block = 128 threads = 4 waves; block computes 16 rows x 128 cols.
// Wave w owns column tiles [32w,32w+16) and [32w+16,32w+32).
// Uses V_WMMA_F32_16X16X4_F32, K swept in steps of 4 (32 WMMA per tile).
// ---------------------------------------------------------------------------
__global__ __launch_bounds__(128)
void gemm_scale(const float* __restrict__ X, const float* __restrict__ W,
                const float* __restrict__ dinv, float* __restrict__ P) {
    __shared__ float xs[16 * 132];            // 132-float pitch: bank-conflict-free
    const int rowBase = blockIdx.x * 16;
    const int tid  = threadIdx.x;

    // Stage 16x128 X tile into LDS (512 float4 chunks, 4 per thread)
    const float4* src4 = reinterpret_cast<const float4*>(X + (size_t)rowBase * DIM);
    float4* xs4 = reinterpret_cast<float4*>(xs);
    #pragma unroll
    for (int i = 0; i < 4; ++i) {
        int c = tid + 128 * i;                // 0..511
        int r = c >> 5;                       // row 0..15 (32 float4 per row)
        int cv = c & 31;                      // float4 col within row
        xs4[r * 33 + cv] = src4[c];           // 33 float4 = 132 float pitch
    }
    __syncthreads();

    const int wave = tid >> 5;
    const int lane = tid & 31;
    const int half = lane >> 4;               // 0: lanes 0-15, 1: lanes 16-31
    const int l16  = lane & 15;
    const int col0 = wave * 32;

    v8f acc0 = {};
    v8f acc1 = {};

    #pragma unroll 4
    for (int k = 0; k < DIM; k += 4) {
        const int kb = k + 2 * half;
        // A frag: 16x4, lanes hold row = l16, VGPR0/1 hold K = kb, kb+1
        v2f a;
        a.x = xs[l16 * 132 + kb];
        a.y = xs[l16 * 132 + kb + 1];
        // B frags: 4x16, lanes hold col = l16, VGPR0/1 hold K = kb, kb+1
        v2f b0, b1;
        b0.x = W[(size_t)kb * DIM + col0 + l16];
        b0.y = W[(size_t)(kb + 1) * DIM + col0 + l16];
        b1.x = W[(size_t)kb * DIM + col0 + 16 + l16];
        b1.y = W[(size_t)(kb + 1) * DIM + col0 + 16 + l16];

        acc0 = __builtin_amdgcn_wmma_f32_16x16x4_f32(
                   false, a, false, b0, (short)0, acc0, false, false);
        acc1 = __builtin_amdgcn_wmma_f32_16x16x4_f32(
                   false, a, false, b1, (short)0, acc1, false, false);
    }

    // Epilogue: D layout — VGPR r: lanes 0-15 -> M=r, lanes 16-31 -> M=8+r
    #pragma unroll
    for (int r = 0; r < 8; ++r) {
        int row = rowBase + r + 8 * half;
        float s = dinv[row];
        P[(size_t)row * DIM + col0 + l16]      = acc0[r] * s;
        P[(size_t)row * DIM + col0 + 16 + l16] = acc1[r] * s;
    }
}

// ---------------------------------------------------------------------------
// Edge scatter: agg[dst] += p[src]   (one wave per edge, 4 feats/lane)
// ---------------------------------------------------------------------------
__global__ __launch_bounds__(256)
void scatter_edges(const int* __restrict__ src, const int* __restrict__ dst,
                   const float* __restrict__ P, float* __restrict__ agg, int e) {
    int gid  = blockIdx.x * blockDim.x + threadIdx.x;
    int edge = gid >> 5;
    int lane = gid & 31;
    if (edge >= e) return;
    int s = src[edge];
    int d = dst[edge];
    float4 v = reinterpret_cast<const float4*>(P + (size_t)s * DIM)[lane];
    float* b = agg + (size_t)d * DIM + lane * 4;
    atomicAdd(b + 0, v.x);
    atomicAdd(b + 1, v.y);
    atomicAdd(b + 2, v.z);
    atomicAdd(b + 3, v.w);
}

// ---------------------------------------------------------------------------
// Finalize: H = act( (agg + p) * dinv[node] + bias )    (p term = self loop)
// ---------------------------------------------------------------------------
__global__ __launch_bounds__(256)
void finalize_layer(const float* __restrict__ agg, const float* __restrict__ P,
                    const float* __restrict__ dinv, const float* __restrict__ bias,
                    float* __restrict__ H, int n, int doRelu) {
    int gid  = blockIdx.x * blockDim.x + threadIdx.x;
    int node = gid >> 5;
    int q    = gid & 31;
    if (node >= n) return;
    float s = dinv[node];
    size_t idx = (size_t)node * DIM + q * 4;
    float4 a = *reinterpret_cast<const float4*>(agg + idx);
    float4 p = *reinterpret_cast<const float4*>(P + idx);
    float4 b = *reinterpret_cast<const float4*>(bias + q * 4);
    float4 o;
    o.x = (a.x + p.x) * s + b.x;
    o.y = (a.y + p.y) * s + b.y;
    o.z = (a.z + p.z) * s + b.z;
    o.w = (a.w + p.w) * s + b.w;
    if (doRelu) {
        o.x = fmaxf(o.x, 0.0f); o.y = fmaxf(o.y, 0.0f);
        o.z = fmaxf(o.z, 0.0f); o.w = fmaxf(o.w, 0.0f);
    }
    *reinterpret_cast<float4*>(H + idx) = o;
}

// ---------------------------------------------------------------------------
// Pool: per-graph feature sums + counts via atomics
// ---------------------------------------------------------------------------
__global__ __launch_bounds__(256)
void pool_sum(const float* __restrict__ H, const int* __restrict__ batch,
              float* __restrict__ gsum, float* __restrict__ gcnt, int n) {
    int gid  = blockIdx.x * blockDim.x + threadIdx.x;
    int node = gid >> 5;
    int q    = gid & 31;
    if (node >= n) return;
    int g = batch[node];
    float4 v = *reinterpret_cast<const float4*>(H + (size_t)node * DIM + q * 4);
    float* b = gsum + (size_t)g * DIM + q * 4;
    atomicAdd(b + 0, v.x);
    atomicAdd(b + 1, v.y);
    atomicAdd(b + 2, v.z);
    atomicAdd(b + 3, v.w);
    if (q == 0) atomicAdd(&gcnt[g], 1.0f);
}

// ---------------------------------------------------------------------------
// Classifier head: out[g] = sigmoid( mean_g . Wc + bc )
// ---------------------------------------------------------------------------
__global__ __launch_bounds__(128)
void classify(const float* __restrict__ gsum, const float* __restrict__ gcnt,
              const float* __restrict__ Wc, const float* __restrict__ bc,
              float* __restrict__ out) {
    __shared__ float red[128];
    int g = blockIdx.x;
    int f = threadIdx.x;
    float cnt = fmaxf(gcnt[g], 1.0f);
    red[f] = (gsum[(size_t)g * DIM + f] / cnt) * Wc[f];
    __syncthreads();
    for (int s = 64; s > 0; s >>= 1) {
        if (f < s) red[f] += red[f + s];
        __syncthreads();
    }
    if (f == 0) {
        float z = red[0] + bc[0];
        out[g] = 1.0f / (1.0f + expf(-z));
    }
}

// ---------------------------------------------------------------------------
// Host-side orchestration (graph-capture safe: only async ops on `stream`)
// ---------------------------------------------------------------------------
extern "C" void kernel_launch(void* const* d_in, const int* in_sizes, int n_in,
                              void* d_out, int out_size, void* d_ws, size_t ws_size,
                              hipStream_t stream) {
    const float* x    = (const float*)d_in[0];
    const int*   ei   = (const int*)  d_in[1];   // [2, E]
    const int*   bat  = (const int*)  d_in[2];
    const float* W1   = (const float*)d_in[3];
    const float* b1   = (const float*)d_in[4];
    const float* W2   = (const float*)d_in[5];
    const float* b2   = (const float*)d_in[6];
    const float* Wc   = (const float*)d_in[7];
    const float* bc   = (const float*)d_in[8];
    float* out = (float*)d_out;

    const int* esrc = ei;
    const int* edst = ei + N_EDGES;

    // workspace layout (floats)
    float* ws    = (float*)d_ws;
    float* dinv  = ws;                                   // 50048 (padded)
    float* p     = ws + 50048;                           // N*128
    float* agg   = p  + (size_t)N_NODES * DIM;           // N*128
    float* h     = agg + (size_t)N_NODES * DIM;          // N*128
    float* gsum  = h  + (size_t)N_NODES * DIM;           // 64*128
    float* gcnt  = gsum + (size_t)NUM_G * DIM;           // 64

    const int TB = 256;
    dim3 nodeGrid((N_NODES + TB - 1) / TB);
    dim3 edgeGrid((N_EDGES + TB - 1) / TB);
    dim3 scatGrid(((size_t)N_EDGES * 32 + TB - 1) / TB);
    dim3 featGrid(((size_t)N_NODES * 32 + TB - 1) / TB);
    dim3 gemmGrid(N_NODES / 16);                         // 50000/16 = 3125 exact

    // --- degree normalization ---
    deg_init <<<nodeGrid, TB, 0, stream>>>(dinv, N_NODES);
    deg_count<<<edgeGrid, TB, 0, stream>>>(edst, dinv, N_EDGES);
    deg_rsqrt<<<nodeGrid, TB, 0, stream>>>(dinv, N_NODES);

    // --- layer 1 ---
    gemm_scale<<<gemmGrid, 128, 0, stream>>>(x, W1, dinv, p);
    hipMemsetAsync(agg, 0, (size_t)N_NODES * DIM * sizeof(float), stream);
    scatter_edges <<<scatGrid, TB, 0, stream>>>(esrc, edst, p, agg, N_EDGES);
    finalize_layer<<<featGrid, TB, 0, stream>>>(agg, p, dinv, b1, h, N_NODES, 1);

    // --- layer 2 ---
    gemm_scale<<<gemmGrid, 128, 0, stream>>>(h, W2, dinv, p);
    hipMemsetAsync(agg, 0, (size_t)N_NODES * DIM * sizeof(float), stream);
    scatter_edges <<<scatGrid, TB, 0, stream>>>(esrc, edst, p, agg, N_EDGES);
    finalize_layer<<<featGrid, TB, 0, stream>>>(agg, p, dinv, b2, h, N_NODES, 0);

    // --- pooling + head ---
    hipMemsetAsync(gsum, 0, ((size_t)NUM_G * DIM + NUM_G) * sizeof(float), stream);
    pool_sum<<<featGrid, TB, 0, stream>>>(h, bat, gsum, gcnt, N_NODES);
    classify<<<NUM_G, DIM, 0, stream>>>(gsum, gcnt, Wc, bc, out);
}